// Model_Aberration_50525995270335
// MI455X (gfx1250) — compile-verified
//
#include <hip/hip_runtime.h>
#include <cstdint>
#include <cmath>

// ---------------------------------------------------------------------------
// Problem constants (from the reference)
// ---------------------------------------------------------------------------
#define NUM_RAY     64
#define NUM_RAY2    (NUM_RAY * NUM_RAY)      // 4096 rays
#define NUM_SAMPLE  64
#define KGRID       64                        // refraction grid edge
#define PH_N        192                       // phantom edge
#define STEP_F      (1.0f / (NUM_SAMPLE - 1)) // 1/63

struct Mats {
    float crd[16];  // row-major; row-vector convention: t[j] = sum_i v[i]*M[i][j]
    float drv[16];
};

// ---------------------------------------------------------------------------
// Kernel 1: ray tracing.  One thread per ray, 63 sequential Euler steps.
// Emits meshXY[r][s][2] = {X*2-1, Y*2-1}  (r-major, s contiguous: 512B/ray).
// ---------------------------------------------------------------------------
__global__ __launch_bounds__(256)
void ray_trace_kernel(const float* __restrict__ RI,       // [K^3][2] (A, sig)
                      const float* __restrict__ rayInit,  // [4096][5]
                      float* __restrict__ meshXY,         // [4096][64][2]
                      Mats M)
{
    const int r = blockIdx.x * blockDim.x + threadIdx.x;
    if (r >= NUM_RAY2) return;

    const float4 init4 = *(const float4*)(rayInit + r * 5);
    float X   = init4.x;
    float dX1 = init4.y;
    float Y   = init4.z;
    float dY1 = init4.w;
    float z   = rayInit[r * 5 + 4];

    float2* outLine = (float2*)(meshXY + r * NUM_SAMPLE * 2);
    outLine[0] = make_float2(X * 2.0f - 1.0f, Y * 2.0f - 1.0f);  // sample 0

    const float* c  = M.crd;
    const float* dv = M.drv;
    const float inv63 = 1.0f / 63.0f;

    for (int s = 0; s < NUM_SAMPLE - 1; ++s) {
        // t = [Y, X, z, 1] @ M_CRD ; then x = t[1], y = t[0], zz = t[2]
        const float t0 = Y * c[0] + X * c[4] + z * c[8]  + c[12];
        const float t1 = Y * c[1] + X * c[5] + z * c[9]  + c[13];
        const float t2 = Y * c[2] + X * c[6] + z * c[10] + c[14];
        const float px = t1, py = t0, pz = t2;

        // 27-neighbor Gaussian mixture (XC[lin] = Xn/63 analytically)
        const float bx = rintf(px * (float)KGRID);
        const float by = rintf(py * (float)KGRID);
        const float bz = rintf(pz * (float)KGRID);

        float norm = 0.f, nus = 0.f;
        float ax = 0.f, ay = 0.f, az = 0.f;     // nu-weighted grads
        float bx_ = 0.f, by_ = 0.f, bz_ = 0.f;  // ni-weighted grads

        #pragma unroll
        for (int ddz = -1; ddz <= 1; ++ddz)
        #pragma unroll
        for (int ddy = -1; ddy <= 1; ++ddy)
        #pragma unroll
        for (int ddx = -1; ddx <= 1; ++ddx) {
            const float Xn = fminf(fmaxf(bx + (float)ddx, 0.f), 63.f);
            const float Yn = fminf(fmaxf(by + (float)ddy, 0.f), 63.f);
            const float Zn = fminf(fmaxf(bz + (float)ddz, 0.f), 63.f);
            const int xi = (int)Xn, yi = (int)Yn, zi = (int)Zn;
            const int lin = (yi << 12) + (xi << 6) + zi;   // 4096*Y + 64*X + Z

            const float dxv = Xn * inv63 - px;
            const float dyv = Yn * inv63 - py;
            const float dzv = Zn * inv63 - pz;

            const float2 ri = *(const float2*)(RI + 2 * lin); // (A, sig)
            const float inv_s2 = 1.0f / (ri.y * ri.y);
            const float r2 = dxv * dxv + dyv * dyv + dzv * dzv;
            const float ni = expf(-r2 * 0.5f * inv_s2) + 2e-7f;
            const float nu = ni * ri.x;

            norm += ni;  nus += nu;
            ax  += nu * dxv * inv_s2;
            ay  += nu * dyv * inv_s2;
            az  += nu * dzv * inv_s2;
            bx_ += ni * dxv * inv_s2;
            by_ += ni * dyv * inv_s2;
            bz_ += ni * dzv * inv_s2;
        }

        const float n = nus / norm;
        const float inv_norm2 = 1.0f / (norm * norm);
        const float gx = (norm * ax - nus * bx_) * inv_norm2;  // dndx_
        const float gy = (norm * ay - nus * by_) * inv_norm2;  // dndy_
        const float gz = (norm * az - nus * bz_) * inv_norm2;  // dndz_

        // d = [dndy_, dndx_, dndz_, 1] @ M_DRV ; dndx = d[1], dndz = d[2], dndy = d[0]
        const float d0 = gy * dv[0] + gx * dv[4] + gz * dv[8]  + dv[12];
        const float d1 = gy * dv[1] + gx * dv[5] + gz * dv[9]  + dv[13];
        const float d2 = gy * dv[2] + gx * dv[6] + gz * dv[10] + dv[14];

        const float dX2 = (d1 - d2 * dX1) * (1.0f + dX1 * dX1) / n;
        const float dY2 = (d0 - d2 * dY1) * (1.0f + dY1 * dY1) / n;

        const float Xn_ = X + dX1 * STEP_F;    // uses old slope (reference order)
        const float Yn_ = Y + dY1 * STEP_F;
        dX1 += dX2 * STEP_F;
        dY1 += dY2 * STEP_F;
        X = Xn_;  Y = Yn_;  z += STEP_F;

        outLine[s + 1] = make_float2(X * 2.0f - 1.0f, Y * 2.0f - 1.0f);
    }
}

// ---------------------------------------------------------------------------
// Kernel 2: fused trilinear resize (64^3 -> 192^3) + grid_sample of Phantom.
// One block per output (i,j) column; 192 threads = one thread per k.
// The 4 corner mesh s-lines (4 x 512B, data-independent addresses) are bulk
// staged into LDS with GLOBAL_LOAD_ASYNC_TO_LDS_B128 (gfx1250 async path).
// ---------------------------------------------------------------------------
__global__ __launch_bounds__(192)
void resample_kernel(const float* __restrict__ phantom,   // [192][192][192]
                     const float* __restrict__ meshXY,    // [4096][64][2]
                     float* __restrict__ out)             // [192][192][192]
{
    const int j   = blockIdx.x;     // b axis of mesh
    const int i   = blockIdx.y;     // a axis of mesh
    const int tid = threadIdx.x;    // k axis (sample axis)

    __shared__ float lines[4 * NUM_SAMPLE * 2];   // [line][s][{X,Y}] = 2 KB

    const float scl = 63.0f / 191.0f;
    const float pa = (float)i * scl;
    const float pb = (float)j * scl;
    const int   a0 = min((int)floorf(pa), 62);
    const int   b0 = min((int)floorf(pb), 62);
    const float wa = pa - (float)a0;
    const float wb = pb - (float)b0;

    // --- async stage: waves 0..3 each copy one 512B s-line into LDS --------
    if (tid < 128) {                                  // whole waves only
        const int l    = tid >> 5;                    // line 0..3 (da*2+db)
        const int lane = tid & 31;
        const int da = l >> 1, db = l & 1;
        const int ray = (a0 + da) * NUM_RAY + (b0 + db);
        const unsigned goff  = (unsigned)(ray * (NUM_SAMPLE * 2 * 4) + lane * 16);
        const unsigned laddr = (unsigned)(uintptr_t)(&lines[0])
                             + (unsigned)(l * 512 + lane * 16);
        asm volatile("global_load_async_to_lds_b128 %0, %1, %2"
                     :
                     : "v"(laddr), "v"(goff), "s"(meshXY)
                     : "memory");
        asm volatile("s_wait_asynccnt 0" ::: "memory");
    }
    __syncthreads();

    // --- per-thread: s-lerp then (a,b) bilinear from LDS -------------------
    const int   k  = tid;
    const float ps = (float)k * scl;
    const int   s0 = min((int)floorf(ps), 62);
    const float ws  = ps - (float)s0;
    const float ws1 = 1.0f - ws;
    const int   o0 = s0 * 2, o1 = (s0 + 1) * 2;

    const float X00 = lines[0 * 128 + o0 + 0] * ws1 + lines[0 * 128 + o1 + 0] * ws;
    const float Y00 = lines[0 * 128 + o0 + 1] * ws1 + lines[0 * 128 + o1 + 1] * ws;
    const float X01 = lines[1 * 128 + o0 + 0] * ws1 + lines[1 * 128 + o1 + 0] * ws;
    const float Y01 = lines[1 * 128 + o0 + 1] * ws1 + lines[1 * 128 + o1 + 1] * ws;
    const float X10 = lines[2 * 128 + o0 + 0] * ws1 + lines[2 * 128 + o1 + 0] * ws;
    const float Y10 = lines[2 * 128 + o0 + 1] * ws1 + lines[2 * 128 + o1 + 1] * ws;
    const float X11 = lines[3 * 128 + o0 + 0] * ws1 + lines[3 * 128 + o1 + 0] * ws;
    const float Y11 = lines[3 * 128 + o0 + 1] * ws1 + lines[3 * 128 + o1 + 1] * ws;

    const float wb1 = 1.0f - wb, wa1 = 1.0f - wa;
    const float gx = (X00 * wb1 + X01 * wb) * wa1 + (X10 * wb1 + X11 * wb) * wa;
    const float gy = (Y00 * wb1 + Y01 * wb) * wa1 + (Y10 * wb1 + Y11 * wb) * wa;
    // z channel is analytic: mesh[2][a,b,s] = 2*s/63 - 1, lerped in s
    const float gz = (2.0f * (float)s0 / 63.0f - 1.0f) * ws1
                   + (2.0f * (float)(s0 + 1) / 63.0f - 1.0f) * ws;

    // --- grid_sample_3d of phantom (border: zero) --------------------------
    const float ix = (gx + 1.0f) * 0.5f * (float)(PH_N - 1);
    const float iy = (gy + 1.0f) * 0.5f * (float)(PH_N - 1);
    const float iz = (gz + 1.0f) * 0.5f * (float)(PH_N - 1);
    const float x0f = floorf(ix), y0f = floorf(iy), z0f = floorf(iz);
    const float fx = ix - x0f, fy = iy - y0f, fz = iz - z0f;

    float acc = 0.0f;
    #pragma unroll
    for (int dzt = 0; dzt <= 1; ++dzt)
    #pragma unroll
    for (int dyt = 0; dyt <= 1; ++dyt)
    #pragma unroll
    for (int dxt = 0; dxt <= 1; ++dxt) {
        const float xf = x0f + (float)dxt;
        const float yf = y0f + (float)dyt;
        const float zf = z0f + (float)dzt;
        const float w = (dxt ? fx : 1.0f - fx)
                      * (dyt ? fy : 1.0f - fy)
                      * (dzt ? fz : 1.0f - fz);
        const bool inb = (xf >= 0.0f) && (xf <= 191.0f) &&
                         (yf >= 0.0f) && (yf <= 191.0f) &&
                         (zf >= 0.0f) && (zf <= 191.0f);
        const int xc = (int)fminf(fmaxf(xf, 0.0f), 191.0f);
        const int yc = (int)fminf(fmaxf(yf, 0.0f), 191.0f);
        const int zc = (int)fminf(fmaxf(zf, 0.0f), 191.0f);
        if (inb)
            acc += w * phantom[(zc * PH_N + yc) * PH_N + xc];
    }

    out[((i * PH_N) + j) * PH_N + k] = acc;   // coalesced along k
}

// ---------------------------------------------------------------------------
// Host: build the 4x4 transforms (exactly as the reference) and launch.
// ---------------------------------------------------------------------------
static void mm4(const double* A, const double* B, double* C) {
    for (int r = 0; r < 4; ++r)
        for (int cc = 0; cc < 4; ++cc) {
            double s = 0.0;
            for (int kk = 0; kk < 4; ++kk) s += A[r * 4 + kk] * B[kk * 4 + cc];
            C[r * 4 + cc] = s;
        }
}
static void mat_ry(double t, double* M) {  // matches reference ry()
    const double c = std::cos(t), s = std::sin(t);
    const double v[16] = {1,0,0,0, 0,c,-s,0, 0,s,c,0, 0,0,0,1};
    for (int q = 0; q < 16; ++q) M[q] = v[q];
}
static void mat_rx(double t, double* M) {  // matches reference rx()
    const double c = std::cos(t), s = std::sin(t);
    const double v[16] = {c,0,-s,0, 0,1,0,0, s,0,c,0, 0,0,0,1};
    for (int q = 0; q < 16; ++q) M[q] = v[q];
}
static void mat_rz(double t, double* M) {  // matches reference rz()
    const double c = std::cos(t), s = std::sin(t);
    const double v[16] = {c,-s,0,0, s,c,0,0, 0,0,1,0, 0,0,0,1};
    for (int q = 0; q < 16; ++q) M[q] = v[q];
}

extern "C" void kernel_launch(void* const* d_in, const int* in_sizes, int n_in,
                              void* d_out, int out_size, void* d_ws, size_t ws_size,
                              hipStream_t stream) {
    (void)in_sizes; (void)n_in; (void)out_size; (void)ws_size;
    const float* phantom = (const float*)d_in[0];   // 192^3
    const float* RI      = (const float*)d_in[1];   // 64^3 * 2
    const float* rayInit = (const float*)d_in[2];   // 4096 * 5
    float* out    = (float*)d_out;                  // 192^3
    float* meshXY = (float*)d_ws;                   // needs 4096*64*2*4 = 2 MB

    // Build M_CRD = S1 @ RZ(-tz) @ RY(-ty) @ RX(tx) @ S2 ; M_DRV = RX(-tx)@RY(ty)@RZ(tz)
    const double THETA_X = 0.1, THETA_Y = 0.05, THETA_Z = 0.02, C = 0.5;
    double RY[16], RY_[16], RX[16], RX_[16], RZ[16], RZ_[16];
    mat_ry(-THETA_Y, RY);  mat_ry(THETA_Y, RY_);
    mat_rx(THETA_X, RX);   mat_rx(-THETA_X, RX_);
    mat_rz(-THETA_Z, RZ);  mat_rz(THETA_Z, RZ_);
    double S1[16] = {1,0,0,0, 0,1,0,0, 0,0,1,0, -C,-C,-C,1};
    double S2[16] = {1,0,0,0, 0,1,0,0, 0,0,1,0,  C, C, C,1};
    double t1[16], t2[16], t3[16], Mcrd[16], Mdrv[16];
    mm4(S1, RZ, t1);  mm4(t1, RY, t2);  mm4(t2, RX, t3);  mm4(t3, S2, Mcrd);
    mm4(RX_, RY_, t1); mm4(t1, RZ_, Mdrv);

    Mats M;
    for (int q = 0; q < 16; ++q) { M.crd[q] = (float)Mcrd[q]; M.drv[q] = (float)Mdrv[q]; }

    ray_trace_kernel<<<NUM_RAY2 / 256, 256, 0, stream>>>(RI, rayInit, meshXY, M);
    resample_kernel<<<dim3(PH_N, PH_N), PH_N, 0, stream>>>(phantom, meshXY, out);
}